// Extractor_60464549593390
// MI455X (gfx1250) — compile-verified
//
#include <hip/hip_runtime.h>
#include <hip/hip_bf16.h>
#include <math.h>

// ---------------------------------------------------------------------------
// MI455X (gfx1250) deformable-attention encoder layer.
// ~45 GFLOP of f32 GEMMs vs ~150MB HBM traffic @ 23.3 TB/s -> memory bound;
// V_WMMA_F32_16X16X4_F32 keeps exact f32 numerics at zero throughput cost.
// All GEMM dims have M%32==0, K%16==0; only N in {48,24} needs bounds checks,
// so the hot GEMMs use unchecked b128 global loads + b128 LDS stores.
// ---------------------------------------------------------------------------

typedef __attribute__((ext_vector_type(2))) float v2f;
typedef __attribute__((ext_vector_type(8))) float v8f;

#define Bsz 4
#define Himg 32
#define Wimg 32
#define Cdim 768
#define NHEAD 6
#define DHEAD 128
#define NPTS 4
#define NQ 5376          // 21*(H*W)/4
#define MQ (Bsz*NQ)      // 21504
#define HWTOK (Himg*Wimg)
#define MF (Bsz*HWTOK)   // 4096
#define HID 192
#define LNEPS 1e-6f

// LDS tile strides (floats): multiples of 4 -> 16B-aligned b128 stores;
// odd multiples of 4 vs bank count checked for conflict-free fragment reads.
#define ASTRIDE 20       // 80B row stride
#define BSTRIDE 36       // 144B row stride

// ---------------------------- LayerNorm ------------------------------------
__launch_bounds__(256)
__global__ void layernorm_k(const float* __restrict__ X, const float* __restrict__ w,
                            const float* __restrict__ b, float* __restrict__ Y, int C)
{
    __shared__ float red[8];
    const int row  = blockIdx.x;
    const int tid  = threadIdx.x;
    const float* x = X + (size_t)row * C;
    float*       y = Y + (size_t)row * C;

    float s = 0.f;
    for (int c = tid; c < C; c += 256) s += x[c];
    for (int o = 16; o; o >>= 1) s += __shfl_xor(s, o, 32);
    if ((tid & 31) == 0) red[tid >> 5] = s;
    __syncthreads();
    float tot = 0.f;
#pragma unroll
    for (int i = 0; i < 8; ++i) tot += red[i];
    const float mu = tot / (float)C;
    __syncthreads();

    float v = 0.f;
    for (int c = tid; c < C; c += 256) { float d = x[c] - mu; v += d * d; }
    for (int o = 16; o; o >>= 1) v += __shfl_xor(v, o, 32);
    if ((tid & 31) == 0) red[tid >> 5] = v;
    __syncthreads();
    float vtot = 0.f;
#pragma unroll
    for (int i = 0; i < 8; ++i) vtot += red[i];
    const float rstd = rsqrtf(vtot / (float)C + LNEPS);

    for (int c = tid; c < C; c += 256)
        y[c] = (x[c] - mu) * rstd * w[c] + b[c];
}

// ------------------------ WMMA f32 GEMM ------------------------------------
// C[M,N] = A[M,K] @ B[K,N] + bias  (+ residual / fused head-transpose)
// 128 threads = 4 waves; block tile 32x32, K-tile 16; LDS-staged.
// CHECK=false: M%32==0, K%16==0, N%32==0 guaranteed -> no predication.
enum { EPI_BIAS = 0, EPI_RES = 1, EPI_VALUE = 2 };

template <int EPI, bool CHECK>
__launch_bounds__(128)
__global__ void gemm_wmma_f32(const float* __restrict__ A,
                              const float* __restrict__ Bm,
                              const float* __restrict__ bias,
                              const float* __restrict__ Res,
                              float* __restrict__ C,
                              int M, int N, int K)
{
    __shared__ float As[32][ASTRIDE];
    __shared__ float Bs[16][BSTRIDE];

    const int tid    = threadIdx.x;
    const int lane   = tid & 31;
    const int wid    = tid >> 5;          // 0..3
    const int wm     = wid >> 1;          // wave row tile (0..1)
    const int wn     = wid & 1;           // wave col tile (0..1)
    const int ntiles = (N + 31) / 32;
    const int m0     = (blockIdx.x / ntiles) * 32;
    const int n0     = (blockIdx.x % ntiles) * 32;
    const int half   = lane >> 4;         // WMMA lane-half
    const int l16    = lane & 15;

    // b128 staging coordinates: A 32x16 (thread -> row tid/4, col 4*(tid%4)),
    //                           B 16x32 (thread -> row tid/8, col 4*(tid%8))
    const int ar = tid >> 2, ac = (tid & 3) << 2;
    const int br = tid >> 3, bc = (tid & 7) << 2;

    v8f acc = {};

    for (int k0 = 0; k0 < K; k0 += 16) {
        // A rows/cols always in range for every launch in this layer
        const float4 av = *(const float4*)(A + (size_t)(m0 + ar) * K + k0 + ac);
        *(float4*)&As[ar][ac] = av;

        if (!CHECK) {
            const float4 bv = *(const float4*)(Bm + (size_t)(k0 + br) * N + n0 + bc);
            *(float4*)&Bs[br][bc] = bv;
        } else {
            const size_t rowb = (size_t)(k0 + br) * N;
#pragma unroll
            for (int j = 0; j < 4; ++j) {
                const int col = n0 + bc + j;
                Bs[br][bc + j] = (col < N) ? Bm[rowb + col] : 0.f;
            }
        }

        // pull next K-tile toward L2/L0 while this one computes
        if (k0 + 16 < K) {
            __builtin_prefetch(A + (size_t)(m0 + ar) * K + k0 + 16 + ac, 0, 1);
            __builtin_prefetch(Bm + (size_t)(k0 + 16 + br) * N + n0 + bc, 0, 1);
        }
        __syncthreads();

#pragma unroll
        for (int kk = 0; kk < 16; kk += 4) {
            // A 16x4 per-lane layout: lanes 0-15 K={kk,kk+1}, lanes 16-31 K={kk+2,kk+3}
            v2f a, b;
            a.x = As[wm * 16 + l16][kk + half * 2 + 0];
            a.y = As[wm * 16 + l16][kk + half * 2 + 1];
            // B 4x16 per-lane layout mirrors A (rows striped across lane halves)
            b.x = Bs[kk + half * 2 + 0][wn * 16 + l16];
            b.y = Bs[kk + half * 2 + 1][wn * 16 + l16];
            acc = __builtin_amdgcn_wmma_f32_16x16x4_f32(
                false, a, false, b, (short)0, acc, false, false);
        }
        __syncthreads();
    }

    // C/D layout: VGPR r -> M = r + half*8, N = l16 (within 16x16 wave tile)
#pragma unroll
    for (int r = 0; r < 8; ++r) {
        const int row = m0 + wm * 16 + half * 8 + r;
        const int col = n0 + wn * 16 + l16;
        if (!CHECK || col < N) {
            float v = acc[r] + bias[col];
            if (EPI == EPI_RES) v += Res[(size_t)row * N + col];
            if (EPI == EPI_VALUE) {
                // fused (B,HW,C) -> (B,Hd,HW,D) transpose for the value tensor
                int bb = row >> 10, l = row & (HWTOK - 1);
                int h  = col >> 7,  d = col & (DHEAD - 1);
                C[((size_t)((bb * NHEAD + h) * HWTOK + l)) * DHEAD + d] = v;
            } else {
                C[(size_t)row * N + col] = v;
            }
        }
    }
}

// ----------------- softmax + bilinear sampling (per-wave task) --------------
// one wave per (b, query, head); lane covers 4 channels via float4 gathers.
__launch_bounds__(256)
__global__ void msdeform_sample_k(const float* __restrict__ value,  // (B,6,1024,128)
                                  const float* __restrict__ off,    // (B,NQ,48)
                                  const float* __restrict__ awl,    // (B,NQ,24)
                                  const float* __restrict__ ref,    // (B,NQ,1,2)
                                  float* __restrict__ out)          // (B,NQ,768)
{
    const int task = blockIdx.x * 8 + (threadIdx.x >> 5);
    const int lane = threadIdx.x & 31;
    if (task >= Bsz * NQ * NHEAD) return;   // grid is exact; never divergent

    const int h  = task % NHEAD;
    const int t  = task / NHEAD;
    const int nq = t % NQ;
    const int b  = t / NQ;

    const size_t qrow = (size_t)(b * NQ + nq);
    const float rx = ref[qrow * 2 + 0];
    const float ry = ref[qrow * 2 + 1];

    // softmax over the 4 points (redundant across lanes, trivially cheap)
    const float* al = awl + qrow * (NHEAD * NPTS) + h * NPTS;
    float l0 = al[0], l1 = al[1], l2 = al[2], l3 = al[3];
    float mx = fmaxf(fmaxf(l0, l1), fmaxf(l2, l3));
    float e0 = expf(l0 - mx), e1 = expf(l1 - mx), e2 = expf(l2 - mx), e3 = expf(l3 - mx);
    float inv = 1.f / (e0 + e1 + e2 + e3);
    float wp[NPTS] = { e0 * inv, e1 * inv, e2 * inv, e3 * inv };

    const float* ofp   = off + qrow * (NHEAD * NPTS * 2) + h * (NPTS * 2);
    const float* vbase = value + (size_t)(b * NHEAD + h) * HWTOK * DHEAD;
    const int d0 = lane * 4;

    float4 accv = make_float4(0.f, 0.f, 0.f, 0.f);
#pragma unroll
    for (int p = 0; p < NPTS; ++p) {
        // pixel coords, grid_sample align_corners=False semantics
        const float lx = rx * (float)Wimg + ofp[p * 2 + 0] - 0.5f;
        const float ly = ry * (float)Himg + ofp[p * 2 + 1] - 0.5f;
        const float x0f = floorf(lx), y0f = floorf(ly);
        const float fx = lx - x0f, fy = ly - y0f;
        const int x0 = (int)x0f, y0 = (int)y0f;
        const float wscale = wp[p];
#pragma unroll
        for (int dy = 0; dy <= 1; ++dy) {
#pragma unroll
            for (int dx = 0; dx <= 1; ++dx) {
                const int xi = x0 + dx, yi = y0 + dy;
                if (xi >= 0 && xi < Wimg && yi >= 0 && yi < Himg) {
                    const float wc = (dx ? fx : 1.f - fx) * (dy ? fy : 1.f - fy);
                    const float4 vv = *(const float4*)(vbase + (size_t)(yi * Wimg + xi) * DHEAD + d0);
                    const float ww = wscale * wc;
                    accv.x += ww * vv.x; accv.y += ww * vv.y;
                    accv.z += ww * vv.z; accv.w += ww * vv.w;
                }
            }
        }
    }
    *(float4*)(out + qrow * Cdim + h * DHEAD + d0) = accv;
}

// ---------------- depthwise 3x3 (3 pyramid branches) + exact GELU -----------
__launch_bounds__(HID)
__global__ void dwconv_gelu_k(const float* __restrict__ X,   // (B,5376,192)
                              const float* __restrict__ Kd,  // (192,1,3,3)
                              const float* __restrict__ bd,  // (192)
                              float* __restrict__ Y)         // (B,5376,192)
{
    const int tok = blockIdx.x % NQ;
    const int b   = blockIdx.x / NQ;
    const int c   = threadIdx.x;      // 0..191, coalesced channel dim

    int base, hgt, wdt, local;
    if (tok < 4096)      { base = 0;    hgt = 64; wdt = 64; local = tok;        }
    else if (tok < 5120) { base = 4096; hgt = 32; wdt = 32; local = tok - 4096; }
    else                 { base = 5120; hgt = 16; wdt = 16; local = tok - 5120; }
    const int y = local / wdt, x = local % wdt;

    const float* k9 = Kd + c * 9;
    float acc = bd[c];
#pragma unroll
    for (int ky = -1; ky <= 1; ++ky) {
        const int yy = y + ky;
        if (yy < 0 || yy >= hgt) continue;
#pragma unroll
        for (int kx = -1; kx <= 1; ++kx) {
            const int xx = x + kx;
            if (xx < 0 || xx >= wdt) continue;
            acc += k9[(ky + 1) * 3 + (kx + 1)] *
                   X[((size_t)b * NQ + base + yy * wdt + xx) * HID + c];
        }
    }
    // exact erf GELU
    const float g = 0.5f * acc * (1.f + erff(acc * 0.70710678118654752f));
    Y[((size_t)b * NQ + tok) * HID + c] = g;
}

// ---------------------------------------------------------------------------
static inline int gemm_grid(int M, int N) { return ((M + 31) / 32) * ((N + 31) / 32); }

extern "C" void kernel_launch(void* const* d_in, const int* in_sizes, int n_in,
                              void* d_out, int out_size, void* d_ws, size_t ws_size,
                              hipStream_t stream) {
    const float* query    = (const float*)d_in[0];
    const float* refpts   = (const float*)d_in[1];
    const float* feat     = (const float*)d_in[2];
    const float* qn_w     = (const float*)d_in[7];
    const float* qn_b     = (const float*)d_in[8];
    const float* fn_w     = (const float*)d_in[9];
    const float* fn_b     = (const float*)d_in[10];
    const float* off_w    = (const float*)d_in[11];
    const float* off_b    = (const float*)d_in[12];
    const float* aw_w     = (const float*)d_in[13];
    const float* aw_b     = (const float*)d_in[14];
    const float* val_w    = (const float*)d_in[15];
    const float* val_b    = (const float*)d_in[16];
    const float* out_w    = (const float*)d_in[17];
    const float* out_b    = (const float*)d_in[18];
    const float* ffn_ln_w = (const float*)d_in[19];
    const float* ffn_ln_b = (const float*)d_in[20];
    const float* fc1_w    = (const float*)d_in[21];
    const float* fc1_b    = (const float*)d_in[22];
    const float* dw_k     = (const float*)d_in[23];
    const float* dw_b     = (const float*)d_in[24];
    const float* fc2_w    = (const float*)d_in[25];
    const float* fc2_b    = (const float*)d_in[26];
    float* out = (float*)d_out;
    float* ws  = (float*)d_ws;

    // workspace layout (floats); buffers reused so peak stays L2-resident-ish
    const size_t O_QN  = 0;                      // qn -> attn_in -> xln -> gelu
    const size_t O_FN  = (size_t)MQ * Cdim;      // fn -> fc1 hidden
    const size_t O_VAL = O_FN  + (size_t)MF * Cdim;
    const size_t O_OFF = O_VAL + (size_t)MF * Cdim;
    const size_t O_AW  = O_OFF + (size_t)MQ * (NHEAD * NPTS * 2);

    float* qn     = ws + O_QN;
    float* fn     = ws + O_FN;
    float* valbuf = ws + O_VAL;
    float* offbuf = ws + O_OFF;
    float* awbuf  = ws + O_AW;
    float* attn   = ws + O_QN;   // overwrites qn after off/aw GEMMs
    float* xln    = ws + O_QN;   // overwrites attn after out-proj
    float* hidden = ws + O_FN;   // overwrites fn/value after sampling
    float* gelu   = ws + O_QN;   // overwrites xln after fc1

    // 1. LayerNorms
    layernorm_k<<<MQ, 256, 0, stream>>>(query, qn_w, qn_b, qn, Cdim);
    layernorm_k<<<MF, 256, 0, stream>>>(feat,  fn_w, fn_b, fn, Cdim);

    // 2. value projection with fused (B,HW,C)->(B,Hd,HW,D) transpose
    gemm_wmma_f32<EPI_VALUE, false><<<gemm_grid(MF, Cdim), 128, 0, stream>>>(
        fn, val_w, val_b, nullptr, valbuf, MF, Cdim, Cdim);

    // 3. offset / attention-weight projections (partial N tiles -> CHECK)
    gemm_wmma_f32<EPI_BIAS, true><<<gemm_grid(MQ, NHEAD * NPTS * 2), 128, 0, stream>>>(
        qn, off_w, off_b, nullptr, offbuf, MQ, NHEAD * NPTS * 2, Cdim);
    gemm_wmma_f32<EPI_BIAS, true><<<gemm_grid(MQ, NHEAD * NPTS), 128, 0, stream>>>(
        qn, aw_w, aw_b, nullptr, awbuf, MQ, NHEAD * NPTS, Cdim);

    // 4. softmax + bilinear sampling (one wave per (b,query,head))
    msdeform_sample_k<<<(Bsz * NQ * NHEAD) / 8, 256, 0, stream>>>(
        valbuf, offbuf, awbuf, refpts, attn);

    // 5. output projection + identity residual -> q (stored in d_out)
    gemm_wmma_f32<EPI_RES, false><<<gemm_grid(MQ, Cdim), 128, 0, stream>>>(
        attn, out_w, out_b, query, out, MQ, Cdim, Cdim);

    // 6. FFN: LN -> fc1 -> dwconv+GELU -> fc2 (+q residual, in place on d_out)
    layernorm_k<<<MQ, 256, 0, stream>>>(out, ffn_ln_w, ffn_ln_b, xln, Cdim);
    gemm_wmma_f32<EPI_BIAS, false><<<gemm_grid(MQ, HID), 128, 0, stream>>>(
        xln, fc1_w, fc1_b, nullptr, hidden, MQ, HID, Cdim);
    dwconv_gelu_k<<<Bsz * NQ, HID, 0, stream>>>(hidden, dw_k, dw_b, gelu);
    gemm_wmma_f32<EPI_RES, false><<<gemm_grid(MQ, Cdim), 128, 0, stream>>>(
        gelu, fc2_w, fc2_b, out, out, MQ, Cdim, HID);
}